// Model_11888469475846
// MI455X (gfx1250) — compile-verified
//
#include <hip/hip_runtime.h>
#include <math.h>
#include <stdint.h>

// Problem constants (from reference)
#define BB   512
#define CC   3
#define HH   64
#define WW   32
#define HOUT 38
#define WOUT 16
#define HID  128
#define DD   114          // C*HOUT
#define DPAD 116          // D padded to multiple of 4 for K=4 WMMA steps
#define TT   16           // sequence length (WOUT)

typedef __attribute__((ext_vector_type(2))) float v2f;   // fp32 WMMA A/B frag (16x4 / 4x16)
typedef __attribute__((ext_vector_type(4))) float v4f;   // native 128-bit vector
typedef __attribute__((ext_vector_type(8))) float v8f;   // fp32 WMMA C/D frag (16x16)

__device__ __forceinline__ float gelu_exact(float v) {
    // exact erf-based GELU (torch nn.GELU default)
    return 0.5f * v * (1.0f + erff(v * 0.70710678118654752f));
}

// ---------------------------------------------------------------------------
// Kernel 1: fractional max-pool (k=2, deterministic u=0.5) + GELU.
// Writes seq[t][b][0..115] (pad lanes 114/115 zeroed) and s_mean[b].
// ROW[i] = floor((2i+1)*31/37) for i<37, ROW[37]=62 ; COL[i] = 2i (exact).
// ---------------------------------------------------------------------------
__global__ void __launch_bounds__(256)
pool_gelu_kernel(const float* __restrict__ x,
                 float* __restrict__ seq,
                 float* __restrict__ s_mean) {
    const int b   = blockIdx.x;
    const int tid = threadIdx.x;
    __shared__ float red[256];

    float lsum = 0.f;
    for (int e = tid; e < CC * HOUT * WOUT; e += 256) {
        const int wo = e & (WOUT - 1);
        const int ho = (e >> 4) % HOUT;
        const int c  = e / (WOUT * HOUT);
        const int r  = (ho < HOUT - 1) ? (((2 * ho + 1) * 31) / 37) : (HH - 2);
        const int col = wo * 2;   // even -> 8B-aligned float2 loads
        const float* p = x + (((b * CC + c) * HH + r) * WW + col);
        const float2 r0 = *(const float2*)(p);
        const float2 r1 = *(const float2*)(p + WW);
        const float m = fmaxf(fmaxf(r0.x, r0.y), fmaxf(r1.x, r1.y));
        lsum += m;
        const int d = c * HOUT + ho;                   // C-major flatten
        seq[((wo * BB) + b) * DPAD + d] = gelu_exact(m);
    }
    // zero the K-padding lanes so WMMA K-tail contributes nothing
    if (tid < WOUT * 2) {
        const int wo = tid >> 1, pdx = tid & 1;
        seq[((wo * BB) + b) * DPAD + DD + pdx] = 0.f;
    }
    red[tid] = lsum;
    __syncthreads();
    for (int s = 128; s > 0; s >>= 1) {
        if (tid < s) red[tid] += red[tid + s];
        __syncthreads();
    }
    if (tid == 0) s_mean[b] = red[0] * (1.0f / (CC * HOUT * WOUT));
}

// ---------------------------------------------------------------------------
// Kernel 2: RNNCell over T=16 steps with V_WMMA_F32_16X16X4_F32.
// 32 workgroups x 16 batch rows; 8 waves/WG, wave j owns cols 16j..16j+15.
// B-matrix fragments (W_ih^T / W_hh^T column slices) preloaded to VGPRs,
// hidden state + X tile staged in LDS, A-fragments gathered per ISA layout:
//   lane l: M = l%16 ; VGPR0 holds K = (l<16 ? 0 : 2)+4k, VGPR1 holds K+1.
// X tile (contiguous 7424 B in global) is staged with the CDNA5 async
// global->LDS path (ASYNCcnt) instead of a VGPR round-trip.
// ---------------------------------------------------------------------------
__global__ void __launch_bounds__(256)
rnn_wmma_kernel(const float* __restrict__ seq,
                const float* __restrict__ W_ih,
                const float* __restrict__ W_hh,
                const float* __restrict__ b_ih,
                const float* __restrict__ b_hh,
                float* __restrict__ t_mean) {
    __shared__ __align__(16) float Xs[16 * DPAD];   // 7.4 KB : timestep input tile
    __shared__ __align__(16) float Hs[16 * HID];    // 8.0 KB : hidden state tile

    const int tid  = threadIdx.x;
    const int wave = tid >> 5;
    const int lane = tid & 31;
    const int row0 = blockIdx.x * 16;         // batch-row tile base
    const int nloc = lane & 15;
    const int n    = wave * 16 + nloc;        // global output column (0..127)
    const int ksel = (lane >> 4) << 1;        // 0 for lanes 0-15, 2 for 16-31
    const int m    = nloc;                    // A-matrix row this lane feeds

    // --- preload B fragments: B[k][n] = W[n][k] (W^T), K padded with zeros ---
    v2f bih[29];
#pragma unroll
    for (int kk = 0; kk < 28; ++kk) {
        const float2 w = *(const float2*)(W_ih + n * DD + kk * 4 + ksel);
        bih[kk].x = w.x; bih[kk].y = w.y;
    }
    if (ksel == 0) {  // kk=28 -> K=112,113 valid only for low half; 114/115 pad
        const float2 w = *(const float2*)(W_ih + n * DD + 112);
        bih[28].x = w.x; bih[28].y = w.y;
    } else {
        bih[28].x = 0.f; bih[28].y = 0.f;
    }
    v2f bhh[32];
#pragma unroll
    for (int kk = 0; kk < 32; ++kk) {
        const float2 w = *(const float2*)(W_hh + n * HID + kk * 4 + ksel);
        bhh[kk].x = w.x; bhh[kk].y = w.y;
    }
    const float bias = b_ih[n] + b_hh[n];

    // h0 = 0
    for (int e = tid; e < 16 * HID; e += 256) Hs[e] = 0.f;

    float hsum[8];
#pragma unroll
    for (int r = 0; r < 8; ++r) hsum[r] = 0.f;

    // per-lane chunk ids for the async X-tile copy: 464 x 16B chunks total
    const int NCHUNK = 16 * (DPAD / 4);                 // 464
    const int e0 = tid;
    int e1 = tid + 256; if (e1 >= NCHUNK) e1 = NCHUNK - 1;  // clamp: dup copy benign
    const uint32_t ldsX = (uint32_t)(uintptr_t)Xs;      // LDS byte offset (addr[31:0])
    const uint32_t lds0 = ldsX + (uint32_t)e0 * 16u;
    const uint32_t lds1 = ldsX + (uint32_t)e1 * 16u;
    const uint32_t g0   = (uint32_t)e0 * 16u;
    const uint32_t g1   = (uint32_t)e1 * 16u;

    for (int t = 0; t < TT; ++t) {
        __syncthreads();   // prev iter done reading Xs / writing Hs

        // async global->LDS copy of the contiguous 16x116 fp32 tile
        {
            const float* gbase = seq + (size_t)(t * BB + row0) * DPAD;
            asm volatile("global_load_async_to_lds_b128 %0, %1, %2"
                         :: "v"(lds0), "v"(g0), "s"(gbase) : "memory");
            asm volatile("global_load_async_to_lds_b128 %0, %1, %2"
                         :: "v"(lds1), "v"(g1), "s"(gbase) : "memory");
            // warm L2/WGP$ for the next timestep's tile while we compute
            if (t + 1 < TT)
                __builtin_prefetch(gbase + (size_t)BB * DPAD + e0 * 4, 0, 3);
            asm volatile("s_wait_asynccnt 0x0" ::: "memory");
        }
        __syncthreads();   // Xs ready (all waves' async copies complete)

        v8f acc;
#pragma unroll
        for (int r = 0; r < 8; ++r) acc[r] = bias;   // b_ih + b_hh (per column)

        // X_t @ W_ih^T  (K = 116, 29 fp32 WMMA k-steps)
#pragma unroll
        for (int kk = 0; kk < 29; ++kk) {
            const int k0 = kk * 4 + ksel;
            v2f a; a.x = Xs[m * DPAD + k0]; a.y = Xs[m * DPAD + k0 + 1];
            acc = __builtin_amdgcn_wmma_f32_16x16x4_f32(
                false, a, false, bih[kk], (short)0, acc, false, false);
        }
        // H @ W_hh^T  (K = 128, 32 fp32 WMMA k-steps)
#pragma unroll
        for (int kk = 0; kk < 32; ++kk) {
            const int k0 = kk * 4 + ksel;
            v2f a; a.x = Hs[m * HID + k0]; a.y = Hs[m * HID + k0 + 1];
            acc = __builtin_amdgcn_wmma_f32_16x16x4_f32(
                false, a, false, bhh[kk], (short)0, acc, false, false);
        }
        __syncthreads();   // everyone done reading Hs before overwrite

        // h = tanh(acc); D-frag layout: VGPR r -> M=r (lanes 0-15) / r+8
#pragma unroll
        for (int r = 0; r < 8; ++r) {
            const float v = tanhf(acc[r]);
            const int rowm = (lane < 16) ? r : (r + 8);
            Hs[rowm * HID + n] = v;
            hsum[r] += v;
        }
    }

    // t_mean = tanh(mean_t(h_t))
#pragma unroll
    for (int r = 0; r < 8; ++r) {
        const int rowm = (lane < 16) ? r : (r + 8);
        t_mean[(row0 + rowm) * HID + n] = tanhf(hsum[r] * (1.0f / TT));
    }
}

// ---------------------------------------------------------------------------
// Kernel 3: out[b0][0][b1][h] = t_mean[b1][h] * s_mean[b0]  (134 MB stream).
// Pure bandwidth: 128-bit non-temporal stores; t_mean/s_mean are L2-resident.
// ---------------------------------------------------------------------------
__global__ void __launch_bounds__(256)
broadcast_kernel(const float* __restrict__ t_mean,
                 const float* __restrict__ s_mean,
                 float* __restrict__ out) {
    const long long i4 = (long long)blockIdx.x * 256 + threadIdx.x;
    const long long i  = i4 * 4;                       // flat fp32 index
    const int b0    = (int)(i >> 16);                  // 512*128 = 65536 per b0
    const int inner = (int)(i & 65535);
    const float s = s_mean[b0];
    v4f tv = *(const v4f*)(t_mean + inner);
    tv *= s;
    __builtin_nontemporal_store(tv, (v4f*)(out + i));
}

// ---------------------------------------------------------------------------
extern "C" void kernel_launch(void* const* d_in, const int* in_sizes, int n_in,
                              void* d_out, int out_size, void* d_ws, size_t ws_size,
                              hipStream_t stream) {
    const float* x    = (const float*)d_in[0];
    const float* W_ih = (const float*)d_in[1];
    const float* W_hh = (const float*)d_in[2];
    const float* b_ih = (const float*)d_in[3];
    const float* b_hh = (const float*)d_in[4];
    float* out = (float*)d_out;

    // workspace layout (fp32): seq[16][512][116] | s_mean[512] | t_mean[512*128]
    float* seq    = (float*)d_ws;
    float* s_mean = seq + (size_t)TT * BB * DPAD;
    float* t_mean = s_mean + BB;

    pool_gelu_kernel<<<BB, 256, 0, stream>>>(x, seq, s_mean);
    rnn_wmma_kernel<<<BB / 16, 256, 0, stream>>>(seq, W_ih, W_hh, b_ih, b_hh, t_mean);

    const long long total4 = (long long)BB * BB * HID / 4;   // 8,388,608 float4
    broadcast_kernel<<<(unsigned)(total4 / 256), 256, 0, stream>>>(t_mean, s_mean, out);
}